// TGAT_11510512353910
// MI455X (gfx1250) — compile-verified
//
#include <hip/hip_runtime.h>
#include <math.h>
#include <stdint.h>

// ---------------------------------------------------------------------------
// TGAT pipeline for MI455X (gfx1250, wave32, WMMA fp32, async LDS staging).
// ---------------------------------------------------------------------------

typedef __attribute__((ext_vector_type(2))) float v2f;
typedef __attribute__((ext_vector_type(8))) float v8f;

#define N_NODES 4096
#define N_EDGES 262144

__device__ __forceinline__ v8f wmma_f32(v2f a, v2f b, v8f c) {
  return __builtin_amdgcn_wmma_f32_16x16x4_f32(false, a, false, b, (short)0, c,
                                               false, false);
}

// CDNA5 async copy: global -> LDS, 16 bytes per lane, tracked by ASYNCcnt.
__device__ __forceinline__ void async_b128(void* lds_dst, const void* gsrc) {
  const uint32_t lds_off = (uint32_t)(uintptr_t)lds_dst;  // low 32 bits = LDS addr
  asm volatile("global_load_async_to_lds_b128 %0, %1, off"
               :
               : "v"(lds_off), "v"(gsrc)
               : "memory");
}
#define WAIT_ASYNC(n) asm volatile("s_wait_asynccnt " #n ::: "memory")

__device__ __forceinline__ float half_reduce_max(float v) {
#pragma unroll
  for (int m = 1; m <= 8; m <<= 1) v = fmaxf(v, __shfl_xor(v, m, 32));
  return v;
}
__device__ __forceinline__ float half_reduce_sum(float v) {
#pragma unroll
  for (int m = 1; m <= 8; m <<= 1) v += __shfl_xor(v, m, 32);
  return v;
}

// ---------------------------------------------------------------------------
// Generic fp32 WMMA GEMM: C[M,N] = A[M,K] @ W[N,K]^T + bias (optional relu).
// Weight panel async-staged into LDS with stride 260 (conflict-free: 260 mod
// 64 banks -> 4m pattern, and 260*4B is a multiple of 16B for B128 alignment).
// ---------------------------------------------------------------------------
__global__ void gemm_wmma(const float* __restrict__ A, const float* __restrict__ W,
                          const float* __restrict__ bias, float* __restrict__ C,
                          int M, int K, int N, int relu) {
  __shared__ alignas(16) float wls[16 * 260];
  const int lane = threadIdx.x & 31;
  const int wv = threadIdx.x >> 5;
  const int tileN = blockIdx.y;
  const int tileM = blockIdx.x * 4 + wv;

  const int kq = K >> 2;  // B128 chunks per row
  for (int i = threadIdx.x; i < 16 * kq; i += blockDim.x) {
    const int c = i / kq;
    const int seg = (i - c * kq) * 4;
    async_b128(&wls[c * 260 + seg], W + (size_t)(tileN * 16 + c) * K + seg);
  }
  WAIT_ASYNC(0);
  __syncthreads();

  const int m = lane & 15;
  const int hi2 = (lane >= 16) ? 2 : 0;
  const float* arow = A + (size_t)(tileM * 16 + m) * K;
  const float* wrow = wls + m * 260;

  v8f acc = {};
  for (int k = 0; k < K; k += 4) {
    v2f a, b;
    a.x = arow[k + hi2];
    a.y = arow[k + hi2 + 1];
    b.x = wrow[k + hi2];
    b.y = wrow[k + hi2 + 1];
    acc = wmma_f32(a, b, acc);
  }

  const int rbase = tileM * 16 + ((lane >= 16) ? 8 : 0);
  const int ccol = tileN * 16 + m;
  const float bb = bias ? bias[ccol] : 0.0f;
#pragma unroll
  for (int r = 0; r < 8; ++r) {
    float v = acc[r] + bb;
    if (relu) v = fmaxf(v, 0.0f);
    C[(size_t)(rbase + r) * N + ccol] = v;
  }
}

// ---------------------------------------------------------------------------
// Temporal encoder (per-edge MLP 1->32->64 + LN), weights in LDS.
// ---------------------------------------------------------------------------
__global__ void temporal_kernel(const float* __restrict__ times,
                                const float* __restrict__ w1, const float* __restrict__ b1,
                                const float* __restrict__ w2, const float* __restrict__ b2,
                                const float* __restrict__ g, const float* __restrict__ bln,
                                const float* __restrict__ decay_p,
                                float* __restrict__ tf) {
  __shared__ float sw2[64 * 32];
  __shared__ float sw1[32], sb1[32], sb2[64], sg[64], sb[64];
  for (int i = threadIdx.x; i < 2048; i += blockDim.x) sw2[i] = w2[i];
  for (int i = threadIdx.x; i < 32; i += blockDim.x) { sw1[i] = w1[i]; sb1[i] = b1[i]; }
  for (int i = threadIdx.x; i < 64; i += blockDim.x) { sb2[i] = b2[i]; sg[i] = g[i]; sb[i] = bln[i]; }
  __syncthreads();

  const int e = blockIdx.x * blockDim.x + threadIdx.x;
  const float decay = decay_p[0];
  const float t = times[e];
  const float wt = t * __expf(-fabsf(t) * decay);

  float h[32];
#pragma unroll
  for (int j = 0; j < 32; ++j) h[j] = fmaxf(wt * sw1[j] + sb1[j], 0.0f);

  float o[64];
  float mu = 0.0f;
  for (int c = 0; c < 64; ++c) {
    float a = sb2[c];
#pragma unroll
    for (int j = 0; j < 32; ++j) a += h[j] * sw2[c * 32 + j];
    o[c] = a;
    mu += a;
  }
  mu *= (1.0f / 64.0f);
  float var = 0.0f;
  for (int c = 0; c < 64; ++c) { float d = o[c] - mu; var += d * d; }
  var *= (1.0f / 64.0f);
  const float rs = rsqrtf(var + 1e-5f);
  for (int c = 0; c < 64; ++c) tf[(size_t)e * 64 + c] = (o[c] - mu) * rs * sg[c] + sb[c];
}

// ---------------------------------------------------------------------------
// Flash attention, fp32 WMMA.  4 waves/block share one head; K/V 16x32 tiles
// are double-buffered in LDS via async loads (issue kb+1, wait kb, compute).
// Row stride 36 floats => conflict-free banks and 16B-aligned B128 targets.
// ---------------------------------------------------------------------------
__global__ void attn_kernel(const float* __restrict__ q, const float* __restrict__ k,
                            const float* __restrict__ v, float* __restrict__ ctx) {
  __shared__ alignas(16) float kt[2][16][36];
  __shared__ alignas(16) float vt[2][16][36];
  __shared__ float pl[4][16][17];
  const int lane = threadIdx.x & 31;
  const int wv = threadIdx.x >> 5;
  const int head = blockIdx.x >> 6;          // all 4 waves: same head
  const int qb = (blockIdx.x & 63) * 4 + wv; // distinct 16-query tiles
  const int m = lane & 15;
  const int hi = (lane >= 16) ? 1 : 0;
  const int hi2 = hi * 2;
  const float scale = 0.17677669529663687f;  // 1/sqrt(32)

  // per-thread staging coordinates: 128 threads x 16B cover one 16x32 tile
  const int srow = threadIdx.x >> 3;
  const int scol = (threadIdx.x & 7) * 4;

  // preload Q A-fragments (whole d=32 reduction)
  v2f qa[8];
  const float* qrow = q + (size_t)(qb * 16 + m) * 128 + head * 32;
#pragma unroll
  for (int s = 0; s < 8; ++s) {
    const int kk = s * 4 + hi2;
    qa[s].x = qrow[kk];
    qa[s].y = qrow[kk + 1];
  }

  float mrow[8], lrow[8];
  v8f o0 = {}, o1 = {};
#pragma unroll
  for (int r = 0; r < 8; ++r) { mrow[r] = -1e30f; lrow[r] = 0.0f; }

  // prologue: stage kb=0 into buffer 0
  async_b128(&kt[0][srow][scol], k + (size_t)srow * 128 + head * 32 + scol);
  async_b128(&vt[0][srow][scol], v + (size_t)srow * 128 + head * 32 + scol);

  for (int kb = 0; kb < 256; ++kb) {
    const int cur = kb & 1;
    const int nxt = cur ^ 1;
    __syncthreads();  // all waves done reading buffer `nxt` (iteration kb-1)
    if (kb + 1 < 256) {
      const float* kg = k + (size_t)((kb + 1) * 16 + srow) * 128 + head * 32 + scol;
      const float* vg = v + (size_t)((kb + 1) * 16 + srow) * 128 + head * 32 + scol;
      async_b128(&kt[nxt][srow][scol], kg);
      async_b128(&vt[nxt][srow][scol], vg);
      WAIT_ASYNC(2);  // drain tile kb; tile kb+1 stays in flight
    } else {
      WAIT_ASYNC(0);
    }
    __syncthreads();  // buffer `cur` visible to all waves

    // S = Q K^T from LDS K tile
    v8f sf = {};
    const float* krow = &kt[cur][m][0];
#pragma unroll
    for (int st = 0; st < 8; ++st) {
      const int kk = st * 4 + hi2;
      v2f b;
      b.x = krow[kk];
      b.y = krow[kk + 1];
      sf = wmma_f32(qa[st], b, sf);
    }
    // online softmax per row
#pragma unroll
    for (int r = 0; r < 8; ++r) {
      const float sv = sf[r] * scale;
      const float mx = fmaxf(half_reduce_max(sv), mrow[r]);
      const float alpha = __expf(mrow[r] - mx);
      mrow[r] = mx;
      const float p = __expf(sv - mx);
      lrow[r] = lrow[r] * alpha + half_reduce_sum(p);
      o0[r] *= alpha;
      o1[r] *= alpha;
      sf[r] = p;
    }
    // C-layout P -> LDS -> A-layout fragments
#pragma unroll
    for (int r = 0; r < 8; ++r) pl[wv][r + 8 * hi][m] = sf[r];
    asm volatile("s_wait_dscnt 0" ::: "memory");
    // O += P V from LDS V tile (two 16-col output tiles)
#pragma unroll
    for (int st = 0; st < 4; ++st) {
      const int kk = st * 4 + hi2;
      v2f a;
      a.x = pl[wv][m][kk];
      a.y = pl[wv][m][kk + 1];
      const float* vr0 = &vt[cur][kk][0];
      v2f b0, b1;
      b0.x = vr0[m];      b0.y = vr0[36 + m];
      b1.x = vr0[16 + m]; b1.y = vr0[36 + 16 + m];
      o0 = wmma_f32(a, b0, o0);
      o1 = wmma_f32(a, b1, o1);
    }
  }

  const int rbase = qb * 16 + 8 * hi;
#pragma unroll
  for (int r = 0; r < 8; ++r) {
    const float inv = 1.0f / lrow[r];
    ctx[(size_t)(rbase + r) * 128 + head * 32 + m] = o0[r] * inv;
    ctx[(size_t)(rbase + r) * 128 + head * 32 + 16 + m] = o1[r] * inv;
  }
}

// ---------------------------------------------------------------------------
// Fused edge MLP: rel_in = [x[src], x[tgt], tf] gathered on the fly,
// h = relu(rel_in @ sc_w1^T + b1)[64], score = sigmoid(h . w2 + b2).
// K=320 staged in 5 async LDS chunks, padded stride 68 (conflict-free).
// ---------------------------------------------------------------------------
__global__ void edge_mlp_kernel(const float* __restrict__ x, const int* __restrict__ ei,
                                const float* __restrict__ tf, const float* __restrict__ w1,
                                const float* __restrict__ b1, const float* __restrict__ w2,
                                const float* __restrict__ b2,
                                float* __restrict__ scores_e) {
  __shared__ alignas(16) float wchunk[64 * 68];
  __shared__ float sw2[64];
  __shared__ float sb1[64];
  const int lane = threadIdx.x & 31;
  const int wv = threadIdx.x >> 5;
  const int e0 = (blockIdx.x * 2 + wv) * 16;
  const int m = lane & 15;
  const int hi2 = (lane >= 16) ? 2 : 0;

  if (threadIdx.x < 64) { sw2[threadIdx.x] = w2[threadIdx.x]; sb1[threadIdx.x] = b1[threadIdx.x]; }

  const int e = e0 + m;
  const int s = ei[e];
  const int t = ei[N_EDGES + e];
  const float* xs = x + (size_t)s * 128;
  const float* xt = x + (size_t)t * 128;
  const float* tfe = tf + (size_t)e * 64;

  v8f acc[4] = {};
  for (int kc = 0; kc < 5; ++kc) {
    __syncthreads();  // previous chunk fully consumed
    for (int i = threadIdx.x; i < 1024; i += blockDim.x) {
      const int c = i >> 4;
      const int seg = (i & 15) * 4;
      async_b128(&wchunk[c * 68 + seg], w1 + (size_t)c * 320 + kc * 64 + seg);
    }
    WAIT_ASYNC(0);
    __syncthreads();
    for (int kl = 0; kl < 64; kl += 4) {
      const int kglob = kc * 64 + kl + hi2;  // region uniform across halves
      const float* ap;
      if (kglob < 128)      ap = xs + kglob;
      else if (kglob < 256) ap = xt + (kglob - 128);
      else                  ap = tfe + (kglob - 256);
      v2f a;
      a.x = ap[0];
      a.y = ap[1];
#pragma unroll
      for (int tt = 0; tt < 4; ++tt) {
        v2f b;
        b.x = wchunk[(tt * 16 + m) * 68 + kl + hi2];
        b.y = wchunk[(tt * 16 + m) * 68 + kl + hi2 + 1];
        acc[tt] = wmma_f32(a, b, acc[tt]);
      }
    }
  }

  const float b2v = b2[0];
  float outv[8];
#pragma unroll
  for (int r = 0; r < 8; ++r) {
    float partial = 0.0f;
#pragma unroll
    for (int tt = 0; tt < 4; ++tt) {
      const int c = tt * 16 + m;
      const float hv = fmaxf(acc[tt][r] + sb1[c], 0.0f);
      partial += hv * sw2[c];
    }
    partial = half_reduce_sum(partial);
    outv[r] = 1.0f / (1.0f + __expf(-(partial + b2v)));
  }
  if (m == 0) {
    const int ebase = e0 + ((lane >= 16) ? 8 : 0);
#pragma unroll
    for (int r = 0; r < 8; ++r) scores_e[ebase + r] = outv[r];
  }
}

// ---------------------------------------------------------------------------
// Scatter / elementwise / LN kernels (L2-resident fp32 atomics)
// ---------------------------------------------------------------------------
__global__ void scatter_edge_kernel(const int* __restrict__ ei,
                                    const float* __restrict__ edge_attr,
                                    const float* __restrict__ base,
                                    float* __restrict__ upd) {
  const int e = blockIdx.x;
  const int f = threadIdx.x;
  const float hier = edge_attr[(size_t)e * 17 + 16];
  if (hier <= 0.0f) return;
  const float tfj = edge_attr[(size_t)e * 17 + 5];
  const float wf = hier * (1.0f / (1.0f + __expf(-tfj))) * 0.3f;
  const float wr = hier * 0.1f;
  const int s = ei[e], t = ei[N_EDGES + e];
  atomicAdd(&upd[(size_t)s * 128 + f], base[(size_t)t * 128 + f] * wf);
  atomicAdd(&upd[(size_t)t * 128 + f], base[(size_t)s * 128 + f] * wr);
}

__global__ void scatter_ctx_kernel(const int* __restrict__ ei,
                                   const float* __restrict__ tf,
                                   const float* __restrict__ se,
                                   float* __restrict__ node_ctx) {
  const int e = blockIdx.x;
  const int f = threadIdx.x;
  atomicAdd(&node_ctx[(size_t)ei[e] * 64 + f], tf[(size_t)e * 64 + f] * se[e]);
}

__global__ void combine_kernel(const float* __restrict__ base, const float* __restrict__ upd,
                               const float* __restrict__ node_ctx, float* __restrict__ comb) {
  const int row = blockIdx.x, f = threadIdx.x;
  const float v = (f < 128) ? (base[(size_t)row * 128 + f] + upd[(size_t)row * 128 + f])
                            : node_ctx[(size_t)row * 64 + (f - 128)];
  comb[(size_t)row * 192 + f] = v;
}

__global__ void ln_relu_kernel(const float* __restrict__ pre, const float* __restrict__ g,
                               const float* __restrict__ b, float* __restrict__ out) {
  __shared__ float red[256];
  const int row = blockIdx.x, c = threadIdx.x;
  const float v = pre[(size_t)row * 256 + c];
  red[c] = v;
  __syncthreads();
  for (int s = 128; s > 0; s >>= 1) { if (c < s) red[c] += red[c + s]; __syncthreads(); }
  const float mu = red[0] * (1.0f / 256.0f);
  __syncthreads();
  const float d = v - mu;
  red[c] = d * d;
  __syncthreads();
  for (int s = 128; s > 0; s >>= 1) { if (c < s) red[c] += red[c + s]; __syncthreads(); }
  const float var = red[0] * (1.0f / 256.0f);
  out[(size_t)row * 256 + c] = fmaxf(d * rsqrtf(var + 1e-5f) * g[c] + b[c], 0.0f);
}

// ---------------------------------------------------------------------------
extern "C" void kernel_launch(void* const* d_in, const int* in_sizes, int n_in,
                              void* d_out, int out_size, void* d_ws, size_t ws_size,
                              hipStream_t stream) {
  const float* x          = (const float*)d_in[0];
  const int*   ei         = (const int*)d_in[1];
  const float* etimes     = (const float*)d_in[2];
  const float* eattr      = (const float*)d_in[3];
  const float* te_w1      = (const float*)d_in[4];
  const float* te_b1      = (const float*)d_in[5];
  const float* te_w2      = (const float*)d_in[6];
  const float* te_b2      = (const float*)d_in[7];
  const float* te_ln_g    = (const float*)d_in[8];
  const float* te_ln_b    = (const float*)d_in[9];
  const float* inp_w      = (const float*)d_in[10];
  const float* inp_b      = (const float*)d_in[11];
  const float* wq         = (const float*)d_in[12];
  const float* bq         = (const float*)d_in[13];
  const float* wk         = (const float*)d_in[14];
  const float* bk         = (const float*)d_in[15];
  const float* wvv        = (const float*)d_in[16];
  const float* bv         = (const float*)d_in[17];
  const float* attn_out_w = (const float*)d_in[18];
  const float* attn_out_b = (const float*)d_in[19];
  const float* outp_w     = (const float*)d_in[20];
  const float* outp_b     = (const float*)d_in[21];
  const float* sc_w1      = (const float*)d_in[22];
  const float* sc_b1      = (const float*)d_in[23];
  const float* sc_w2      = (const float*)d_in[24];
  const float* sc_b2      = (const float*)d_in[25];
  const float* agg_w      = (const float*)d_in[26];
  const float* agg_b      = (const float*)d_in[27];
  const float* agg_ln_g   = (const float*)d_in[28];
  const float* agg_ln_b   = (const float*)d_in[29];
  const float* decay      = (const float*)d_in[30];
  float* out = (float*)d_out;

  float* ws = (float*)d_ws;
  float* tf       = ws;                               // 262144*64
  float* xp       = tf + (size_t)N_EDGES * 64;
  float* q        = xp + (size_t)N_NODES * 128;
  float* kbuf     = q + (size_t)N_NODES * 128;
  float* vbuf     = kbuf + (size_t)N_NODES * 128;
  float* ctx      = vbuf + (size_t)N_NODES * 128;
  float* upd      = ctx + (size_t)N_NODES * 128;
  float* scores_e = upd + (size_t)N_NODES * 128;
  float* node_ctx = scores_e + (size_t)N_EDGES;
  float* comb     = node_ctx + (size_t)N_NODES * 64;
  float* attn_out = q;     // q dead after attention
  float* base     = kbuf;  // k dead after attention
  float* pre      = tf;    // tf dead after edge MLP + ctx scatter

  hipMemsetAsync(upd, 0, (size_t)N_NODES * 128 * sizeof(float), stream);
  hipMemsetAsync(node_ctx, 0, (size_t)N_NODES * 64 * sizeof(float), stream);

  temporal_kernel<<<N_EDGES / 64, 64, 0, stream>>>(etimes, te_w1, te_b1, te_w2, te_b2,
                                                   te_ln_g, te_ln_b, decay, tf);
  gemm_wmma<<<dim3(64, 8), 128, 0, stream>>>(x, inp_w, inp_b, xp, N_NODES, 128, 128, 0);
  gemm_wmma<<<dim3(64, 8), 128, 0, stream>>>(xp, wq, bq, q, N_NODES, 128, 128, 0);
  gemm_wmma<<<dim3(64, 8), 128, 0, stream>>>(xp, wk, bk, kbuf, N_NODES, 128, 128, 0);
  gemm_wmma<<<dim3(64, 8), 128, 0, stream>>>(xp, wvv, bv, vbuf, N_NODES, 128, 128, 0);
  attn_kernel<<<256, 128, 0, stream>>>(q, kbuf, vbuf, ctx);
  gemm_wmma<<<dim3(64, 8), 128, 0, stream>>>(ctx, attn_out_w, attn_out_b, attn_out,
                                             N_NODES, 128, 128, 0);
  gemm_wmma<<<dim3(64, 8), 128, 0, stream>>>(attn_out, outp_w, outp_b, base,
                                             N_NODES, 128, 128, 0);
  scatter_edge_kernel<<<N_EDGES, 128, 0, stream>>>(ei, eattr, base, upd);
  edge_mlp_kernel<<<N_EDGES / 32, 64, 0, stream>>>(x, ei, tf, sc_w1, sc_b1, sc_w2,
                                                   sc_b2, scores_e);
  scatter_ctx_kernel<<<N_EDGES, 64, 0, stream>>>(ei, tf, scores_e, node_ctx);
  combine_kernel<<<N_NODES, 192, 0, stream>>>(base, upd, node_ctx, comb);
  gemm_wmma<<<dim3(64, 16), 128, 0, stream>>>(comb, agg_w, agg_b, pre, N_NODES, 192, 256, 0);
  ln_relu_kernel<<<N_NODES, 256, 0, stream>>>(pre, agg_ln_g, agg_ln_b, out);
}